// ProdAt_5411658793348
// MI455X (gfx1250) — compile-verified
//
#include <hip/hip_runtime.h>
#include <stdint.h>

// Segment product: out[o] = prod(x[o*64 .. o*64+63]); x is 16384 x 8192 f32.
// Pure streaming kernel: 512 MiB in / 8 MiB out -> ~23 us floor at 23.3 TB/s.
// Strategy: async global->LDS DMA (gfx1250 ASYNCcnt path) with padded LDS
// layout for conflict-free per-segment reads, double-buffered per block.

#define THREADS           256
#define SEG_LEN_F         64
#define SEG_BYTES         256                         // 64 floats
#define SEG_STRIDE        272                         // 256 + 16B pad: 68 dwords == 4 (mod 64 banks)
#define TILE_SEGS         256                         // segments per tile (1 per thread)
#define TILE_BYTES        (TILE_SEGS * SEG_BYTES)     // 65536
#define BUF_BYTES         (TILE_SEGS * SEG_STRIDE)    // 69632 (x2 buffers = 136 KB LDS)
#define TILES_PER_BLOCK   8
#define CHUNKS_PER_THREAD ((TILE_BYTES / 16) / THREADS)  // 16 async b128 loads / thread / tile

// The builtin's signature (from the round-1 diagnostic) is:
//   (v4i32 addrspace(1)*, v4i32 addrspace(3)*, imm offset, imm cpol)
typedef int v4i __attribute__((vector_size(16)));
typedef __attribute__((address_space(1))) v4i* gv4i_p;   // global <4 x i32>*
typedef __attribute__((address_space(3))) v4i* lv4i_p;   // LDS    <4 x i32>*

// ---- gfx1250 async global->LDS copy (per-lane LDS dest enables padding) ----
__device__ __forceinline__ void async_g2l_b128(const void* g, void* l) {
#if __has_builtin(__builtin_amdgcn_global_load_async_to_lds_b128)
    __builtin_amdgcn_global_load_async_to_lds_b128(
        (gv4i_p)(unsigned long long)g,   // AS1 == flat repr: int cast is safe
        (lv4i_p)l,                       // generic -> AS3: addrspacecast
        /*offset=*/0, /*cpol=*/0);
#elif defined(__HIP_DEVICE_COMPILE__)
    unsigned lds_off      = (unsigned)(size_t)(__attribute__((address_space(3))) char*)l;
    unsigned long long ga = (unsigned long long)g;
    asm volatile("global_load_async_to_lds_b128 %0, %1, off"
                 :: "v"(lds_off), "v"(ga) : "memory");
#else
    (void)g; (void)l;   // host pass: never executed, keep sema trivial
#endif
}

template <int N>
__device__ __forceinline__ void wait_asynccnt() {
#if __has_builtin(__builtin_amdgcn_s_wait_asynccnt)
    __builtin_amdgcn_s_wait_asynccnt(N);
#elif defined(__HIP_DEVICE_COMPILE__)
    asm volatile("s_wait_asynccnt %0" :: "i"(N) : "memory");
#endif
}

__global__ __launch_bounds__(THREADS)
void prodseg_async_kernel(const float* __restrict__ x,
                          float* __restrict__ out,
                          int nTiles)
{
    extern __shared__ char smem[];   // 2 * BUF_BYTES, padded tiles
    const int t = (int)threadIdx.x;

    // Thread t owns 16B chunk c = t + j*256 of the tile.
    // Padded LDS addr(c) = (c>>4)*SEG_STRIDE + (c&15)*16
    //  -> per-thread base + j * (16*SEG_STRIDE), since c&15 == t&15 for all j.
    const unsigned ldsThreadBase =
        (unsigned)((t >> 4) * SEG_STRIDE + (t & 15) * 16);

    const int tile0 = (int)blockIdx.x * TILES_PER_BLOCK;
    if (tile0 >= nTiles) return;

    const char* xb = (const char*)x;

    // Prime the pipeline: prefetch tile0 into buffer 0.
    {
        const char* g = xb + (size_t)tile0 * TILE_BYTES + (size_t)t * 16;
        char*       l = smem + ldsThreadBase;
        #pragma unroll
        for (int j = 0; j < CHUNKS_PER_THREAD; ++j)
            async_g2l_b128(g + (size_t)j * (THREADS * 16),
                           l + j * (16 * SEG_STRIDE));
    }

    int buf = 0;
    for (int i = 0; i < TILES_PER_BLOCK; ++i) {     // block-uniform control flow
        const int tile = tile0 + i;
        if (tile >= nTiles) break;

        const bool more = (i + 1 < TILES_PER_BLOCK) && (tile + 1 < nTiles);
        if (more) {
            // Prefetch next tile into the other buffer, then wait only for
            // the current tile's 16 async loads (ASYNCcnt <= 16).
            const char* g = xb + (size_t)(tile + 1) * TILE_BYTES + (size_t)t * 16;
            char*       l = smem + (buf ^ 1) * BUF_BYTES + ldsThreadBase;
            #pragma unroll
            for (int j = 0; j < CHUNKS_PER_THREAD; ++j)
                async_g2l_b128(g + (size_t)j * (THREADS * 16),
                               l + j * (16 * SEG_STRIDE));
            wait_asynccnt<CHUNKS_PER_THREAD>();
        } else {
            wait_asynccnt<0>();
        }
        __syncthreads();   // all waves' async writes for tile i are visible

        // Product of this thread's segment: 16 contiguous float4 LDS loads
        // (stride 272B -> conflict-free across the wave), multiply tree.
        const float4* q =
            (const float4*)(smem + buf * BUF_BYTES + t * SEG_STRIDE);
        float4 v[16];
        #pragma unroll
        for (int k = 0; k < 16; ++k) v[k] = q[k];
        #pragma unroll
        for (int s2 = 8; s2 >= 1; s2 >>= 1) {
            #pragma unroll
            for (int k = 0; k < s2; ++k) {
                v[k].x *= v[k + s2].x;
                v[k].y *= v[k + s2].y;
                v[k].z *= v[k + s2].z;
                v[k].w *= v[k + s2].w;
            }
        }
        const float p = (v[0].x * v[0].y) * (v[0].z * v[0].w);
        out[(size_t)tile * TILE_SEGS + t] = p;   // contiguous per block

        __syncthreads();   // buffer `buf` free before it is refilled at i+2
        buf ^= 1;
    }
}

extern "C" void kernel_launch(void* const* d_in, const int* in_sizes, int n_in,
                              void* d_out, int out_size, void* d_ws, size_t ws_size,
                              hipStream_t stream)
{
    (void)n_in; (void)out_size; (void)d_ws; (void)ws_size;

    const float* x  = (const float*)d_in[0];
    float*      out = (float*)d_out;

    const long long n      = in_sizes[0];                 // 134,217,728 floats
    const int nSegs        = (int)(n / SEG_LEN_F);        // 2,097,152
    const int nTiles       = (nSegs + TILE_SEGS - 1) / TILE_SEGS;              // 8192
    const int nBlocks      = (nTiles + TILES_PER_BLOCK - 1) / TILES_PER_BLOCK; // 1024
    const size_t shmem     = 2 * BUF_BYTES;               // 139,264 B (<320 KB/WGP)

    // Opt in to >64KB dynamic LDS (CDNA5 WGP supports 320 KB).
    (void)hipFuncSetAttribute((const void*)prodseg_async_kernel,
                              hipFuncAttributeMaxDynamicSharedMemorySize,
                              (int)shmem);

    prodseg_async_kernel<<<nBlocks, THREADS, shmem, stream>>>(x, out, nTiles);
}